// GraphSAGE_40973988004535
// MI455X (gfx1250) — compile-verified
//
#include <hip/hip_runtime.h>

typedef float v2f __attribute__((ext_vector_type(2)));
typedef float v8f __attribute__((ext_vector_type(8)));

#define FEAT 128
#define LDS_STRIDE 132   // 132*4B = 528B row stride -> rows shifted 4 banks apart: conflict-free column reads

// ---------------- zero fill ----------------
__global__ __launch_bounds__(256) void fill_zero_kernel(float* __restrict__ p, long long n) {
    long long i = blockIdx.x * 256LL + threadIdx.x;
    long long j = i * 4;
    if (j + 4 <= n) {
        ((float4*)p)[i] = make_float4(0.f, 0.f, 0.f, 0.f);
    } else if (j < n) {
        for (long long k = j; k < n; ++k) p[k] = 0.f;
    }
}

// ---------------- in-degree ----------------
__global__ __launch_bounds__(256) void degree_kernel(const int* __restrict__ dst,
                                                     float* __restrict__ deg, int n_edges) {
    int e = blockIdx.x * 256 + threadIdx.x;
    if (e < n_edges) atomicAdd(&deg[dst[e]], 1.0f);
}

// ------------- edge scatter-add: agg[dst] += feat[src] -------------
// 32 lanes per edge, float4 per lane -> one coalesced 512B gather per edge,
// atomics land in L2 (accumulator is 51MB, fits the 192MB L2).
__global__ __launch_bounds__(256) void scatter_kernel(const float* __restrict__ feat,
                                                      const int* __restrict__ src,
                                                      const int* __restrict__ dst,
                                                      float* __restrict__ agg, int n_edges) {
    long long tid = blockIdx.x * 256LL + threadIdx.x;
    int e = (int)(tid >> 5);
    if (e >= n_edges) return;
    int lane = (int)(tid & 31);
    long long s = src[e];
    long long d = dst[e];
    float4 v = ((const float4*)(feat + s * FEAT))[lane];
    float* o = agg + d * FEAT + lane * 4;
    atomicAdd(o + 0, v.x);
    atomicAdd(o + 1, v.y);
    atomicAdd(o + 2, v.z);
    atomicAdd(o + 3, v.w);
}

// ------------- fused SAGE layer: OUT = act(X@W0 + (AGG/deg)@W1) -------------
// One block per 16-row node tile; wave w (of 8) computes output cols [16w,16w+16).
// fp32 WMMA (V_WMMA_F32_16X16X4_F32), K loop of 32 x 2 WMMAs.
__global__ __launch_bounds__(256) void sage_gemm_kernel(
        const float* __restrict__ X, const float* __restrict__ AGG,
        const float* __restrict__ DEG,
        const float* __restrict__ W0, const float* __restrict__ W1,
        float* __restrict__ OUT, int n_nodes, int do_relu)
{
    __shared__ float sA[16 * LDS_STRIDE];   // self features tile
    __shared__ float sN[16 * LDS_STRIDE];   // mean-neighbor tile
    const int row0 = blockIdx.x * 16;
    const int t = threadIdx.x;

    // cooperative coalesced load of 16x128 tiles (512 float4s, 2 per thread)
    for (int i = t; i < 512; i += 256) {
        int r  = i >> 5;    // row in tile
        int c4 = i & 31;    // float4 column
        long long node = row0 + r;
        float4 xv = make_float4(0.f, 0.f, 0.f, 0.f);
        float4 av = make_float4(0.f, 0.f, 0.f, 0.f);
        float rd = 0.f;
        if (node < n_nodes) {
            xv = ((const float4*)(X   + node * FEAT))[c4];
            av = ((const float4*)(AGG + node * FEAT))[c4];
            rd = 1.0f / fmaxf(DEG[node], 1.0f);
        }
        ((float4*)&sA[r * LDS_STRIDE])[c4] = xv;
        float4 nv = make_float4(av.x * rd, av.y * rd, av.z * rd, av.w * rd);
        ((float4*)&sN[r * LDS_STRIDE])[c4] = nv;
    }
    __syncthreads();

    const int wave = t >> 5;           // 0..7 -> output column tile
    const int lane = t & 31;
    const int col0 = wave * 16;
    const int m     = lane & 15;       // A: row M / B: col N
    const int khalf = (lane >> 4) << 1; // lanes 16-31 hold K+2,K+3

    v8f acc = {0.f, 0.f, 0.f, 0.f, 0.f, 0.f, 0.f, 0.f};

#pragma unroll
    for (int k0 = 0; k0 < FEAT; k0 += 4) {
        const int ka = k0 + khalf;
        // A fragments from LDS (8B aligned, conflict-free due to padding)
        v2f a  = *(const v2f*)&sA[m * LDS_STRIDE + ka];
        v2f an = *(const v2f*)&sN[m * LDS_STRIDE + ka];
        // B fragments: W[k][n], row-major [128 x 128] (hot in L1/L2)
        v2f b0, b1;
        b0.x = W0[ ka      * FEAT + col0 + m];
        b0.y = W0[(ka + 1) * FEAT + col0 + m];
        b1.x = W1[ ka      * FEAT + col0 + m];
        b1.y = W1[(ka + 1) * FEAT + col0 + m];
        acc = __builtin_amdgcn_wmma_f32_16x16x4_f32(false, a,  false, b0, (short)0, acc, false, false);
        acc = __builtin_amdgcn_wmma_f32_16x16x4_f32(false, an, false, b1, (short)0, acc, false, false);
    }

    // C/D layout: lane<16 -> N=lane, M=vgpr; lane>=16 -> N=lane-16, M=vgpr+8
    const int ncol  = col0 + m;
    const int mbase = (lane >> 4) << 3;
#pragma unroll
    for (int i2 = 0; i2 < 8; ++i2) {
        long long node = row0 + mbase + i2;
        if (node < n_nodes) {
            float v = acc[i2];
            if (do_relu) v = fmaxf(v, 0.f);
            OUT[node * FEAT + ncol] = v;
        }
    }
}

extern "C" void kernel_launch(void* const* d_in, const int* in_sizes, int n_in,
                              void* d_out, int out_size, void* d_ws, size_t ws_size,
                              hipStream_t stream) {
    const float* x     = (const float*)d_in[0];   // [N,128]
    const float* W_in  = (const float*)d_in[1];   // [2,128,128]
    const float* W_out = (const float*)d_in[2];   // [2,128,128]
    const int*   src   = (const int*)d_in[3];     // [E]
    const int*   dst   = (const int*)d_in[4];     // [E]
    float* out = (float*)d_out;

    const int N = in_sizes[0] / FEAT;
    const int E = in_sizes[3];

    // workspace: agg[N*128] | h[N*128] | deg[N]  (~103 MB)
    float* agg = (float*)d_ws;
    float* h   = agg + (size_t)N * FEAT;
    float* deg = h   + (size_t)N * FEAT;

    const long long featElems = (long long)N * FEAT;
    dim3 blk(256);
    dim3 gridFillFeat((unsigned)(((featElems + 3) / 4 + 255) / 256));
    dim3 gridFillDeg((unsigned)((((long long)N + 3) / 4 + 255) / 256));
    dim3 gridDeg((unsigned)((E + 255) / 256));
    dim3 gridScat((unsigned)(((long long)E * 32 + 255) / 256));
    dim3 gridGemm((unsigned)((N + 15) / 16));

    // degree (shared by both layers)
    fill_zero_kernel<<<gridFillDeg, blk, 0, stream>>>(deg, (long long)N);
    degree_kernel<<<gridDeg, blk, 0, stream>>>(dst, deg, E);

    // ---- layer 0 ----
    fill_zero_kernel<<<gridFillFeat, blk, 0, stream>>>(agg, featElems);
    scatter_kernel<<<gridScat, blk, 0, stream>>>(x, src, dst, agg, E);
    sage_gemm_kernel<<<gridGemm, blk, 0, stream>>>(x, agg, deg,
                                                   W_in, W_in + FEAT * FEAT,
                                                   h, N, /*relu=*/1);

    // ---- layer 1 ----
    fill_zero_kernel<<<gridFillFeat, blk, 0, stream>>>(agg, featElems);
    scatter_kernel<<<gridScat, blk, 0, stream>>>(h, src, dst, agg, E);
    sage_gemm_kernel<<<gridGemm, blk, 0, stream>>>(h, agg, deg,
                                                   W_out, W_out + FEAT * FEAT,
                                                   out, N, /*relu=*/0);
}